// GNN_74131135529535
// MI455X (gfx1250) — compile-verified
//
#include <hip/hip_runtime.h>

// CDNA5 / gfx1250, wave32. GEMMs via V_WMMA_F32_16X16X4_F32 (fp32 in/out).
typedef __attribute__((ext_vector_type(2))) float v2f;
typedef __attribute__((ext_vector_type(8))) float v8f;

#define NB   8      // batch
#define KH   3      // hops
#define NN   192    // nodes
#define DOBS 32
#define DIN  64
#define TN   64     // every GEMM in this net has N == 64
#define ITILE 8     // output rows per edge_reduce block

// ---- optional CDNA5 async global->LDS path (ASYNCcnt), with safe fallback ----
#if defined(__HIP_DEVICE_COMPILE__) &&                                  \
    __has_builtin(__builtin_amdgcn_global_load_async_to_lds_b32) &&     \
    __has_builtin(__builtin_amdgcn_s_wait_asynccnt)
#define USE_ASYNC_LDS 1
#else
#define USE_ASYNC_LDS 0
#endif

#if USE_ASYNC_LDS
// Builtin signature (per hipcc diagnostic): (global int*, shared int*, imm, imm)
typedef __attribute__((address_space(1))) int gas_int;
typedef __attribute__((address_space(3))) int las_int;
__device__ __forceinline__ gas_int* to_glb(const void* p) {
  return (gas_int*)(unsigned long long)p;
}
__device__ __forceinline__ las_int* to_lds(const void* p) {
  // generic shared address: low 32 bits are the LDS byte offset (ISA 10.2)
  return (las_int*)(unsigned)(unsigned long long)p;
}
#endif

// ---------------------------------------------------------------------------
// Batched GEMM, N fixed at 64: C[bz] = A[bz] (MxKd) * B[bz] (Kdx64) (+bias)(+relu)
// blockDim = 128 (4 waves). Block computes a 64x64 output tile; wave w owns a
// 16x64 strip (4 WMMA accumulators). B is staged 16 rows at a time in LDS and
// shared by all 4 waves (ds_load feeds the WMMA B operand).
// A 16x4 fp32 operand layout: VGPR0 = {K=0 | K=2}, VGPR1 = {K=1 | K=3} with the
// lane halves selecting; B mirrored; C/D: VGPR v = rows {v | v+8}.
// epi: 0 = none, 1 = +bias, 2 = relu(+bias).
// ---------------------------------------------------------------------------
__global__ void wmma_gemm64(const float* __restrict__ Ab, long sA,
                            const float* __restrict__ Bb, long sB,
                            const float* __restrict__ bias,
                            float* __restrict__ Cb, long sC,
                            int M, int Kd, int lda, int epi) {
  const int bz   = blockIdx.z;
  const int wave = threadIdx.x >> 5;
  const int lane = threadIdx.x & 31;
  const int lm   = lane & 15;          // M row (A) / N col (B) within tile
  const int kh   = (lane >> 4) << 1;   // K sub-offset: 0 or 2
  const int mh   = (lane >> 4) << 3;   // +8 rows for upper lane half
  const int m0   = blockIdx.x * 64 + wave * 16;

  const float* A = Ab + (long)bz * sA + (long)m0 * lda;
  const float* B = Bb + (long)bz * sB;
  float*       C = Cb + (long)bz * sC + (long)m0 * TN;

  __shared__ float sBt[16 * TN];       // one 16x64 B chunk (4 KB)

  v8f acc[4] = {};

  for (int kc = 0; kc < Kd; kc += 16) {
    __syncthreads();                   // protect LDS reuse across chunks
    // stage B[kc..kc+16, 0..64): 1024 floats, 128 threads -> 8 each
    #pragma unroll
    for (int r = 0; r < 8; ++r) {
      const int idx = threadIdx.x + r * 128;
      sBt[idx] = B[(long)(kc + (idx >> 6)) * TN + (idx & 63)];
    }
    __syncthreads();
    #pragma unroll
    for (int k0 = 0; k0 < 16; k0 += 4) {
      v2f a;
      const float* ap = A + (long)lm * lda + (kc + k0 + kh);
      a.x = ap[0];
      a.y = ap[1];
      #pragma unroll
      for (int tn = 0; tn < 4; ++tn) {
        v2f b;
        const float* bp = &sBt[(k0 + kh) * TN + tn * 16 + lm];
        b.x = bp[0];
        b.y = bp[TN];
        acc[tn] = __builtin_amdgcn_wmma_f32_16x16x4_f32(
            false, a, false, b, (short)0, acc[tn], false, false);
      }
    }
  }

  #pragma unroll
  for (int tn = 0; tn < 4; ++tn) {
    const int col = tn * 16 + lm;
    const float bv = (epi != 0) ? bias[col] : 0.0f;
    #pragma unroll
    for (int v = 0; v < 8; ++v) {
      float val = acc[tn][v] + bv;
      if (epi == 2) val = fmaxf(val, 0.0f);
      C[(long)(mh + v) * TN + col] = val;
    }
  }
}

// ---------------------------------------------------------------------------
// agg[b,i,k,:]  k=1..3: sum_j relu(A0[b,i,j] * Pk[b,j,:] + bpre)
// agg[b,i,0,:]       : relu(Q0[b,i,:] + bpre) + (N-1)*relu(bpre)   (diag hop-0)
// 8 output rows per block: each P[b,j,:] element loaded once, used 8x (VALU-
// bound). 8 adjacency rows staged in LDS via async global->LDS when available.
// 256 threads: t = kk*64 + d for the 3 hop channels; t in [192,256) does hop-0.
// ---------------------------------------------------------------------------
__global__ void edge_reduce(const float* __restrict__ A,     // [B,KH,N,N]
                            const float* __restrict__ P,     // [3,B,N,DIN]
                            const float* __restrict__ Q0,    // [B,N,DIN]
                            const float* __restrict__ bpre,  // [DIN]
                            float* __restrict__ agg) {       // [B,N,4,DIN]
  const int i0 = blockIdx.x * ITILE;
  const int b  = blockIdx.y;
  const int t  = threadIdx.x;          // 256

  __shared__ float sA[ITILE][NN];      // 6 KB
  float* sAf = &sA[0][0];
  const float* arow = A + ((long)b * KH) * NN * NN + (long)i0 * NN;

#if USE_ASYNC_LDS
  #pragma unroll
  for (int r = 0; r < (ITILE * NN) / 256; ++r) {
    const int idx = t + r * 256;
    __builtin_amdgcn_global_load_async_to_lds_b32(to_glb(arow + idx),
                                                  to_lds(sAf + idx), 0, 0);
  }
  __builtin_amdgcn_s_wait_asynccnt(0);
#else
  #pragma unroll
  for (int r = 0; r < (ITILE * NN) / 256; ++r) {
    const int idx = t + r * 256;
    sAf[idx] = arow[idx];
  }
#endif
  __syncthreads();

  if (t < 3 * DIN) {
    const int kk = t >> 6;             // hop-1 index 0..2
    const int d  = t & 63;
    const float bp = bpre[d];
    const float* Pk = P + ((long)(kk * NB + b) * NN) * DIN + d;
    float acc[ITILE] = {};
    for (int j = 0; j < NN; ++j) {
      const float p = Pk[(long)j * DIN];
      #pragma unroll
      for (int r = 0; r < ITILE; ++r) {
        acc[r] += fmaxf(fmaf(sA[r][j], p, bp), 0.0f);
      }
    }
    #pragma unroll
    for (int r = 0; r < ITILE; ++r) {
      agg[(((long)b * NN + i0 + r) * 4 + (kk + 1)) * DIN + d] = acc[r];
    }
  } else {
    const int d = t - 3 * DIN;
    const float bp = bpre[d];
    const float rv = (float)(NN - 1) * fmaxf(bp, 0.0f);
    #pragma unroll
    for (int r = 0; r < ITILE; ++r) {
      const float q = Q0[((long)b * NN + i0 + r) * DIN + d];
      agg[(((long)b * NN + i0 + r) * 4) * DIN + d] = fmaxf(q + bp, 0.0f) + rv;
    }
  }
}

// s[r,:] = sum_{k=0..3} m[r,k,:]
__global__ void hop_sum(const float* __restrict__ m, float* __restrict__ s, int total) {
  const int idx = blockIdx.x * blockDim.x + threadIdx.x;
  if (idx >= total) return;
  const int r = idx >> 6, d = idx & 63;
  const float* mp = m + ((long)r * 4) * DIN + d;
  s[idx] = mp[0] + mp[DIN] + mp[2 * DIN] + mp[3 * DIN];
}

extern "C" void kernel_launch(void* const* d_in, const int* in_sizes, int n_in,
                              void* d_out, int out_size, void* d_ws, size_t ws_size,
                              hipStream_t stream) {
  (void)in_sizes; (void)n_in; (void)out_size; (void)ws_size;
  const float* A    = (const float*)d_in[0];  // [8,3,192,192]
  const float* X    = (const float*)d_in[1];  // [8,4,192,32]
  const float* Wi   = (const float*)d_in[2];
  const float* bi   = (const float*)d_in[3];
  const float* Wpre = (const float*)d_in[4];
  const float* bpre = (const float*)d_in[5];
  const float* Wmid = (const float*)d_in[6];
  const float* bmid = (const float*)d_in[7];
  const float* Wout = (const float*)d_in[8];
  const float* bout = (const float*)d_in[9];
  float* out = (float*)d_out;                 // [8,192,64]
  float* ws  = (float*)d_ws;

  const long NNsq = (long)NN * NN;   // 36864
  const long ND   = (long)NN * DIN;  // 12288

  // Workspace layout (floats); peak ~5.9 MB.
  float* Xc   = ws;                          // [B,4,N,DIN]  393216
  float* Y12  = Xc   + (long)NB * 4 * ND;    //  98304
  float* Y11  = Y12  + (long)NB * ND;        //  98304
  float* Y21  = Y11  + (long)NB * ND;        //  98304
  float* Pbuf = Y21  + (long)NB * ND;        // [3,B,N,DIN] 294912
  float* Q0   = Pbuf + 3L * NB * ND;         //  98304
  float* agg  = Q0   + (long)NB * ND;        // [B,N,4,DIN] 393216
  float* mbuf = Xc;    // reuse: Xc dead after projections
  float* sbuf = Y11;   // reuse: Y11 dead after projections

  // 1) Xc = relu(X @ Wi + bi)   [B*4*N, 32] x [32, 64]
  wmma_gemm64<<<dim3(NB * 4 * NN / 64, 1, 1), 128, 0, stream>>>(
      X, 0, Wi, 0, bi, Xc, 0, NB * 4 * NN, DOBS, DOBS, 2);
  // 2) Y11 = A[:,1] @ Xc[:,2]
  wmma_gemm64<<<dim3(NN / 64, 1, NB), 128, 0, stream>>>(
      A + 1 * NNsq, 3 * NNsq, Xc + 2 * ND, 4 * ND, nullptr, Y11, ND,
      NN, NN, NN, 0);
  // 3) Y12 = A[:,2] @ Xc[:,3]
  wmma_gemm64<<<dim3(NN / 64, 1, NB), 128, 0, stream>>>(
      A + 2 * NNsq, 3 * NNsq, Xc + 3 * ND, 4 * ND, nullptr, Y12, ND,
      NN, NN, NN, 0);
  // 4) Y21 = A[:,1] @ Y12
  wmma_gemm64<<<dim3(NN / 64, 1, NB), 128, 0, stream>>>(
      A + 1 * NNsq, 3 * NNsq, Y12, ND, nullptr, Y21, ND, NN, NN, NN, 0);
  // 5) Projections through Wpre (linear, so the per-edge scalar factors out)
  wmma_gemm64<<<dim3(NN / 64, 1, NB), 128, 0, stream>>>(
      Xc + 0 * ND, 4 * ND, Wpre, 0, nullptr, Q0, ND, NN, DIN, DIN, 0);
  wmma_gemm64<<<dim3(NN / 64, 1, NB), 128, 0, stream>>>(
      Xc + 1 * ND, 4 * ND, Wpre, 0, nullptr, Pbuf + 0L * NB * ND, ND,
      NN, DIN, DIN, 0);
  wmma_gemm64<<<dim3(NN / 64, 1, NB), 128, 0, stream>>>(
      Y11, ND, Wpre, 0, nullptr, Pbuf + 1L * NB * ND, ND, NN, DIN, DIN, 0);
  wmma_gemm64<<<dim3(NN / 64, 1, NB), 128, 0, stream>>>(
      Y21, ND, Wpre, 0, nullptr, Pbuf + 2L * NB * ND, ND, NN, DIN, DIN, 0);
  // 6) Per-edge relu reduction -> agg[B,N,4,DIN]
  edge_reduce<<<dim3(NN / ITILE, NB), 256, 0, stream>>>(A, Pbuf, Q0, bpre, agg);
  // 7) m = relu(agg @ Wmid + bmid)   [B*N*4, 64] x [64, 64]
  wmma_gemm64<<<dim3(NB * NN * 4 / 64, 1, 1), 128, 0, stream>>>(
      agg, 0, Wmid, 0, bmid, mbuf, 0, NB * NN * 4, DIN, DIN, 2);
  // 8) s = sum over hops
  hop_sum<<<(NB * NN * DIN + 255) / 256, 256, 0, stream>>>(mbuf, sbuf, NB * NN * DIN);
  // 9) out = s @ Wout + bout
  wmma_gemm64<<<dim3(NB * NN / 64, 1, 1), 128, 0, stream>>>(
      sbuf, 0, Wout, 0, bout, out, 0, NB * NN, DIN, DIN, 1);
}